// MyLoss_75161927680507
// MI455X (gfx1250) — compile-verified
//
#include <hip/hip_runtime.h>

// ---------------------------------------------------------------------------
// MyLoss: weighted log-loss reduction, N = 16,777,216 f32 samples.
// Memory-bound: 8 B/sample (f32 prob + i32 label) = 134 MB @ 23.3 TB/s ≈ 5.8us.
// Design: grid-stride float4/int4 streaming -> branchless per-sample math
// (v_log_f32 + 2x v_cndmask + v_fma) -> per-thread f32 partial -> wave32
// reduction via V_WMMA_F32_16X16X4_F32 (A = partials, B = ones) -> block
// partial in d_ws -> deterministic single-block finalize into d_out[0].
// ---------------------------------------------------------------------------

typedef __attribute__((ext_vector_type(2))) float v2f;
typedef __attribute__((ext_vector_type(8))) float v8f;

__device__ __forceinline__ float wave32_sum(float v) {
#if defined(__gfx1250__) && __has_builtin(__builtin_amdgcn_wmma_f32_16x16x4_f32)
    // A-matrix 16x4 f32 layout (ISA 7.12.2): lanes 0-15 VGPR0 -> (M=lane,K=0),
    // lanes 16-31 VGPR0 -> (M=lane-16,K=2). Put partial in a.x, 0 in a.y,
    // B = ones => D[m][n] = partial(m) + partial(m+16) for every n.
    v2f a; a.x = v;    a.y = 0.0f;
    v2f b; b.x = 1.0f; b.y = 1.0f;
    v8f c = {};
    v8f d = __builtin_amdgcn_wmma_f32_16x16x4_f32(
        /*neg_a=*/false, a, /*neg_b=*/false, b,
        /*c_mod=*/(short)0, c, /*reuse_a=*/false, /*reuse_b=*/false);
    // Lane L<16 holds D[r][L] = pairsum(r), r=0..7; lane L>=16 holds pairsum(r+8).
    float s = ((d[0] + d[1]) + (d[2] + d[3])) + ((d[4] + d[5]) + (d[6] + d[7]));
    s += __shfl_xor(s, 16, 32);   // combine low-M half with high-M half
    return s;
#else
    for (int off = 16; off > 0; off >>= 1) v += __shfl_xor(v, off, 32);
    return v;
#endif
}

__device__ __forceinline__ float sample_loss(float x, int lab, float w0, float w1) {
    // neg_log = -ln(x) = log2(x) * (-ln2)          (v_log_f32 + v_mul_f32)
    // label==1 -> nl*w1 ; label==0 -> (1-nl)*w0 ; else 0
    // Branchless: w = lab==1 ? w1 : 0 ; u = lab==0 ? w0 : 0
    //             contrib = nl*w + (1-nl)*u = fma(nl, w-u, u)
    const float nl = __builtin_amdgcn_logf(x) * (-0.69314718055994531f);
    const float w  = (lab == 1) ? w1 : 0.0f;      // v_cndmask
    const float u  = (lab == 0) ? w0 : 0.0f;      // v_cndmask
    return __builtin_fmaf(nl, w - u, u);          // v_sub + v_fma
}

__device__ __forceinline__ void block_reduce_store(float acc, float* dst) {
    __shared__ float smem[8];                    // up to 8 wave32 per 256-thr block
    const int lane = threadIdx.x & 31;
    const int wid  = threadIdx.x >> 5;
    float ws = wave32_sum(acc);                  // EXEC all-ones here (no divergence)
    if (lane == 0) smem[wid] = ws;
    __syncthreads();
    if (wid == 0) {                              // uniform branch: wave 0 fully active
        float v = (lane < (int)(blockDim.x >> 5)) ? smem[lane] : 0.0f;
        v = wave32_sum(v);
        if (lane == 0) *dst = v;
    }
}

__global__ __launch_bounds__(256) void loss_partial_kernel(
    const float* __restrict__ prob, const int* __restrict__ label,
    const float* __restrict__ weights, float* __restrict__ partials, int n) {
    const float w0 = weights[0];
    const float w1 = weights[1];

    const int gid    = blockIdx.x * blockDim.x + threadIdx.x;
    const int stride = gridDim.x * blockDim.x;
    const int nv     = n >> 2;                    // float4 / int4 chunks

    const float4* __restrict__ p4 = (const float4*)prob;
    const int4*   __restrict__ l4 = (const int4*)label;

    float acc0 = 0.0f, acc1 = 0.0f;               // two chains to hide ALU latency
    for (int i = gid; i < nv; i += stride) {
        const float4 x = p4[i];                   // global_load_b128
        const int4   l = l4[i];                   // global_load_b128
        acc0 += sample_loss(x.x, l.x, w0, w1);
        acc1 += sample_loss(x.y, l.y, w0, w1);
        acc0 += sample_loss(x.z, l.z, w0, w1);
        acc1 += sample_loss(x.w, l.w, w0, w1);
    }
    // Scalar tail (N is a multiple of 4, but stay generic).
    for (int i = (nv << 2) + gid; i < n; i += stride)
        acc0 += sample_loss(prob[i], label[i], w0, w1);

    block_reduce_store(acc0 + acc1, &partials[blockIdx.x]);
}

__global__ __launch_bounds__(256) void loss_finalize_kernel(
    const float* __restrict__ partials, int m, float* __restrict__ out) {
    float acc = 0.0f;
    for (int i = threadIdx.x; i < m; i += (int)blockDim.x) acc += partials[i];
    block_reduce_store(acc, out);
}

extern "C" void kernel_launch(void* const* d_in, const int* in_sizes, int n_in,
                              void* d_out, int out_size, void* d_ws, size_t ws_size,
                              hipStream_t stream) {
    const float* prob    = (const float*)d_in[0];
    const int*   label   = (const int*)d_in[1];
    const float* weights = (const float*)d_in[2];
    const int    n       = in_sizes[0];

    float* partials = (float*)d_ws;

    // 2048 blocks x 256 thr = 524,288 wave32 lanes: 32 samples/thread, enough
    // outstanding b128 loads to saturate 23.3 TB/s. Clamp to scratch capacity.
    int blocks = 2048;
    const size_t max_blocks = ws_size / sizeof(float);
    if ((size_t)blocks > max_blocks && max_blocks > 0) blocks = (int)max_blocks;
    if (blocks < 1) blocks = 1;

    loss_partial_kernel<<<blocks, 256, 0, stream>>>(prob, label, weights, partials, n);
    loss_finalize_kernel<<<1, 256, 0, stream>>>(partials, blocks, (float*)d_out);
}